// SortPooling_65326452572944
// MI455X (gfx1250) — compile-verified
//
#include <hip/hip_runtime.h>
#include <hip/hip_bf16.h>

// Problem constants (from reference): B=32, N=8192, F=256, K=1024
#define B_   32
#define N_   8192
#define F_   256
#define K_   1024
#define BINS 2048     // 11-bit histogram of sortable key
#define MAXC 6144     // candidate cap (typical C ~ 1.4k for N(0,1) keys)

#define ROWS_PB 8     // gather rows per block (8 KB LDS staging)

typedef int v4i __attribute__((ext_vector_type(4)));

#if defined(__gfx1250__)
#if !__has_builtin(__builtin_amdgcn_global_load_async_to_lds_b128) || \
    !__has_builtin(__builtin_amdgcn_global_store_async_from_lds_b128)
#error "gfx1250 async-LDS builtins not available on this toolchain"
#endif
#define USE_ASYNC_LDS 1
typedef __attribute__((address_space(3))) v4i las_v4i;
#endif

static __device__ __forceinline__ void wait_asynccnt0() {
#if defined(__gfx1250__)
#if __has_builtin(__builtin_amdgcn_s_wait_asynccnt)
    __builtin_amdgcn_s_wait_asynccnt(0);
#else
    asm volatile("s_wait_asynccnt 0" ::: "memory");
#endif
#endif
}

// Monotonic float->u32 transform: f0 < f1  <=>  sortable(f0) < sortable(f1)
static __device__ __forceinline__ unsigned sortable(float f) {
    unsigned u = __float_as_uint(f);
    return u ^ ((u & 0x80000000u) ? 0xFFFFFFFFu : 0x80000000u);
}

// ---------------------------------------------------------------------------
// Kernel 1: per-batch exact top-K rank -> source-index table
// grid = B_, block = 1024 (32 waves), dynamic LDS
// ---------------------------------------------------------------------------
__global__ void topk_rank_kernel(const float* __restrict__ x,
                                 int* __restrict__ out_idx) {
    extern __shared__ unsigned char smem_raw[];
    unsigned long long* cand = (unsigned long long*)smem_raw;           // MAXC * 8B
    unsigned* hist = (unsigned*)(smem_raw + (size_t)MAXC * 8);          // BINS
    unsigned* sup  = hist + BINS;                                       // 64 superbins
    unsigned* vars = sup + 64;                                          // [0]=cutbin [1]=candCount

    const int b = blockIdx.x;
    const int t = threadIdx.x;
    const float* xb = x + (size_t)b * N_ * F_;

    for (int i = t; i < BINS; i += 1024) hist[i] = 0;
    if (t == 0) vars[1] = 0;
    __syncthreads();

    // --- histogram of top 11 bits; stash own sortable keys in registers ---
    unsigned myS[N_ / 1024];
#pragma unroll
    for (int c = 0; c < N_ / 1024; ++c) {
        int i = t + c * 1024;
        unsigned s = sortable(xb[(size_t)i * F_ + (F_ - 1)]);
        myS[c] = s;
        atomicAdd(&hist[s >> 21], 1u);
    }
    __syncthreads();

    // --- two-level suffix scan from the top to locate the cut bin ---
    if (t < 64) {
        unsigned acc = 0;
        for (int j = 0; j < 32; ++j) acc += hist[t * 32 + j];
        sup[t] = acc;
    }
    __syncthreads();

    if (t == 0) {
        unsigned cum = 0;
        int sb = 63;
        for (; sb > 0; --sb) {
            if (cum + sup[sb] >= K_) break;
            cum += sup[sb];
        }
        int cutbin = sb * 32;
        for (int bin = sb * 32 + 31; bin >= sb * 32; --bin) {
            if (cum + hist[bin] >= K_) { cutbin = bin; break; }
            cum += hist[bin];
        }
        vars[0] = (unsigned)cutbin;
    }
    __syncthreads();

    // --- compact candidates (bin >= cutbin) as composite 64-bit keys ---
    const unsigned cutbin = vars[0];
#pragma unroll
    for (int c = 0; c < N_ / 1024; ++c) {
        unsigned s = myS[c];
        if ((s >> 21) >= cutbin) {
            unsigned p = atomicAdd(&vars[1], 1u);
            if (p < MAXC) {
                int i = t + c * 1024;
                // larger composite == earlier in descending top_k order
                cand[p] = ((unsigned long long)s << 13) |
                          (unsigned long long)(unsigned)((N_ - 1) - i);
            }
        }
    }
    __syncthreads();

    // --- exact rank among candidates (== global rank); scatter indices ---
    int C = (int)vars[1];
    if (C > MAXC) C = MAXC;
    for (int c = t; c < C; c += 1024) {
        const unsigned long long me = cand[c];
        int r = 0;
        for (int j = 0; j < C; ++j) r += (cand[j] > me) ? 1 : 0;  // LDS broadcast
        if (r < K_) {
            int idx = (N_ - 1) - (int)(me & 0x1FFFull);
            out_idx[b * K_ + r] = idx;
        }
    }
}

// ---------------------------------------------------------------------------
// Kernel 2: gather selected rows, streamed through LDS via async DMA ops
// (global_load_async_to_lds_b128 / global_store_async_from_lds_b128,
//  ASYNCcnt-tracked, drained with s_wait_asynccnt)
// grid = (B_*K_)/ROWS_PB, block = 256
// ---------------------------------------------------------------------------
__global__ void gather_rows_kernel(const float* __restrict__ x,
                                   const int* __restrict__ out_idx,
                                   float* __restrict__ out) {
    __shared__ float buf[ROWS_PB * F_];   // 8 KB staging
    const int row0 = blockIdx.x * ROWS_PB;  // first output row [0, B_*K_)
    const int t = threadIdx.x;

    const int UNITS = ROWS_PB * (F_ / 4);   // 16B transfer units: 512
    for (int u = t; u < UNITS; u += 256) {
        int lr = u >> 6;              // local row (64 units per 1KB row)
        int e  = u & 63;              // float4 index within row
        int grow = row0 + lr;
        int b = grow >> 10;           // / K_
        int r = grow & (K_ - 1);
        int src = out_idx[b * K_ + r];
        const float4* gp = (const float4*)(x + ((size_t)b * N_ + src) * F_) + e;
#ifdef USE_ASYNC_LDS
        __builtin_amdgcn_global_load_async_to_lds_b128(
            (v4i*)gp, (las_v4i*)(buf + (size_t)u * 4), 0, 0);
#else
        ((float4*)buf)[u] = *gp;
#endif
    }
#ifdef USE_ASYNC_LDS
    wait_asynccnt0();
#endif
    __syncthreads();

    float4* op = (float4*)out + (size_t)row0 * (F_ / 4);
    for (int u = t; u < UNITS; u += 256) {
#ifdef USE_ASYNC_LDS
        __builtin_amdgcn_global_store_async_from_lds_b128(
            (v4i*)(op + u), (las_v4i*)(buf + (size_t)u * 4), 0, 0);
#else
        op[u] = ((float4*)buf)[u];
#endif
    }
#ifdef USE_ASYNC_LDS
    wait_asynccnt0();
#endif
}

// ---------------------------------------------------------------------------
extern "C" void kernel_launch(void* const* d_in, const int* in_sizes, int n_in,
                              void* d_out, int out_size, void* d_ws, size_t ws_size,
                              hipStream_t stream) {
    const float* x = (const float*)d_in[0];   // [B, N, F] f32
    // d_in[1] is k (== 1024), compile-time constant here
    float* out = (float*)d_out;               // [B, K, F] f32
    int* idxbuf = (int*)d_ws;                 // B_*K_ ints = 128 KB scratch

    const size_t smem1 = (size_t)MAXC * 8 + (size_t)(BINS + 64 + 8) * 4;
    topk_rank_kernel<<<B_, 1024, smem1, stream>>>(x, idxbuf);

    const int nblocks = (B_ * K_) / ROWS_PB;  // 4096
    gather_rows_kernel<<<nblocks, 256, 0, stream>>>(x, idxbuf, out);
}